// matrix_times_16303695855981
// MI455X (gfx1250) — compile-verified
//
#include <hip/hip_runtime.h>
#include <stdint.h>

typedef __attribute__((ext_vector_type(2))) float v2f;
typedef __attribute__((ext_vector_type(8))) float v8f;

#define DIM 4096
#define BM 128
#define BN 128
#define BK 16
#define LDP 20  // padded LDS row stride (floats): 80B = 16B-aligned, conflict-free frags
#define KTILES (DIM / BK)

// C = A(Jacobian) @ B(Eye), fp32 row-major DIMxDIM.
// Block = 256 threads (8 wave32) -> 128x128 C tile; waves 4(M) x 2(N), each
// wave owns 32x64 => 2x4 fragments, 8 chained V_WMMA_F32_16X16X4_F32 per kk.
// Staging uses CDNA5 async copies (GLOBAL_LOAD_ASYNC_TO_LDS_*, ASYNCcnt):
//   A tile 128x16 -> As[m][k]   via 2 x async b128 per thread
//   B tile 16x128 -> BsT[n][k]  via 8 x async b32 per thread (transpose in-flight)
// Double-buffered; DMA for tile kt+1 overlaps WMMA on tile kt.
__global__ __launch_bounds__(256) void gemm_f32_wmma_async_kernel(
    const float* __restrict__ A, const float* __restrict__ B,
    float* __restrict__ C) {
  __shared__ float As[2][BM][LDP];
  __shared__ float BsT[2][BN][LDP];

  const int tid  = threadIdx.x;
  const int wave = tid >> 5;     // 0..7
  const int lane = tid & 31;
  const int half = lane >> 4;    // 0 or 1
  const int lq   = lane & 15;    // 0..15

  const int wrow = (wave & 3) * 32;  // wave M offset: 0,32,64,96
  const int wcol = (wave >> 2) * 64; // wave N offset: 0,64

  const int block_row = blockIdx.y * BM;
  const int block_col = blockIdx.x * BN;

  // ---- async staging addressing (GVS mode: SGPR64 base + per-lane VGPR32) ----
  // A tile: thread covers rows (arow, arow+64), one 16B chunk at acol.
  const int arow = tid >> 2;          // 0..63
  const int acol = (tid & 3) << 2;    // 0,4,8,12
  const unsigned avoff0 = (unsigned)(((size_t)arow * DIM + acol) * 4);
  const unsigned avoff1 = (unsigned)(((size_t)(arow + 64) * DIM + acol) * 4);

  // B tile (transposed into LDS): thread owns column bn, k rows bkh..bkh+7.
  const int bn  = tid & 127;          // 0..127
  const int bkh = (tid >> 7) << 3;    // 0 or 8
  unsigned bvoff[8];
#pragma unroll
  for (int i = 0; i < 8; ++i)
    bvoff[i] = (unsigned)(((size_t)(bkh + i) * DIM + bn) * 4);

  // LDS destination byte addresses (wave-relative = low 32 bits of generic ptr).
  unsigned ldsA0[2], ldsA1[2], ldsB0[2];
#pragma unroll
  for (int q = 0; q < 2; ++q) {
    ldsA0[q] = (unsigned)(uintptr_t)&As[q][arow][acol];
    ldsA1[q] = (unsigned)(uintptr_t)&As[q][arow + 64][acol];
    ldsB0[q] = (unsigned)(uintptr_t)&BsT[q][bn][bkh];
  }

  // Uniform SGPR bases; only these advance with kt.
  const char* const aBase = (const char*)(A + (size_t)block_row * DIM);
  const char* const bBase = (const char*)(B + (size_t)block_col);

  auto issue_tile = [&](int kt, int q) {
    const char* ga = aBase + (size_t)kt * (BK * 4);            // +64B / K-step
    const char* gb = bBase + (size_t)kt * ((size_t)BK * DIM * 4);
    asm volatile("global_load_async_to_lds_b128 %0, %1, %2"
                 :: "v"(ldsA0[q]), "v"(avoff0), "s"(ga) : "memory");
    asm volatile("global_load_async_to_lds_b128 %0, %1, %2"
                 :: "v"(ldsA1[q]), "v"(avoff1), "s"(ga) : "memory");
#pragma unroll
    for (int i = 0; i < 8; ++i) {
      const unsigned ldsb = ldsB0[q] + (unsigned)(i * 4);
      asm volatile("global_load_async_to_lds_b32 %0, %1, %2"
                   :: "v"(ldsb), "v"(bvoff[i]), "s"(gb) : "memory");
    }
  };

  v8f acc[2][4];
#pragma unroll
  for (int i = 0; i < 2; ++i)
#pragma unroll
    for (int j = 0; j < 4; ++j) acc[i][j] = (v8f)0.0f;

  // Prime buffer 0 with K-tile 0.
  issue_tile(0, 0);
  asm volatile("s_wait_asynccnt 0x0" ::: "memory");
  __syncthreads();

  int p = 0;
  for (int kt = 0; kt < KTILES; ++kt) {
    // Kick off DMA of K-tile kt+1 into the other buffer; it runs under the WMMAs.
    if (kt + 1 < KTILES) issue_tile(kt + 1, p ^ 1);

    // Compute: 4 chained 16x16x4 fp32 WMMAs over this K-tile, 2x4 C fragments.
#pragma unroll
    for (int kk = 0; kk < BK; kk += 4) {
      const int kb = kk + 2 * half;  // lanes 0-15: K=kk,kk+1; lanes 16-31: +2
      v2f af[2];
      v2f bf[4];
#pragma unroll
      for (int i = 0; i < 2; ++i)
        af[i] = *(const v2f*)&As[p][wrow + i * 16 + lq][kb];
#pragma unroll
      for (int j = 0; j < 4; ++j)
        bf[j] = *(const v2f*)&BsT[p][wcol + j * 16 + lq][kb];
#pragma unroll
      for (int i = 0; i < 2; ++i)
#pragma unroll
        for (int j = 0; j < 4; ++j)
          acc[i][j] = __builtin_amdgcn_wmma_f32_16x16x4_f32(
              false, af[i], false, bf[j], (short)0, acc[i][j], false, false);
    }

    // Drain this wave's async copies, then block-wide handoff of the buffer.
    asm volatile("s_wait_asynccnt 0x0" ::: "memory");
    __syncthreads();
    p ^= 1;
  }

  // Write back: C/D layout = 8 VGPRs; VGPR r -> M = r (lanes 0-15) / r+8 (lanes 16-31).
#pragma unroll
  for (int i = 0; i < 2; ++i) {
#pragma unroll
    for (int j = 0; j < 4; ++j) {
      const int col = block_col + wcol + j * 16 + lq;
#pragma unroll
      for (int r = 0; r < 8; ++r) {
        const int row = block_row + wrow + i * 16 + r + 8 * half;
        C[(size_t)row * DIM + col] = acc[i][j][r];
      }
    }
  }
}

extern "C" void kernel_launch(void* const* d_in, const int* in_sizes, int n_in,
                              void* d_out, int out_size, void* d_ws, size_t ws_size,
                              hipStream_t stream) {
  // setup_inputs order: 0=curr_t 1=curr_x 2=dyn_para 3=rand_para 4=eye 5=LE
  //                     6=random_value 7=jacobian
  const float* eye      = (const float*)d_in[4];
  const float* jacobian = (const float*)d_in[7];
  float* out = (float*)d_out;

  dim3 grid(DIM / BN, DIM / BM);  // 32 x 32
  dim3 block(256);
  gemm_f32_wmma_async_kernel<<<grid, block, 0, stream>>>(jacobian, eye, out);
}